// FDC5_ada_61108794688084
// MI455X (gfx1250) — compile-verified
//
#include <hip/hip_runtime.h>

typedef __attribute__((ext_vector_type(2))) float v2f;
typedef __attribute__((ext_vector_type(8))) float v8f;

#define BN_EPS 1e-5f

__global__ __launch_bounds__(256) void zero_f32_kern(float* __restrict__ p, int n) {
  int i = blockIdx.x * blockDim.x + threadIdx.x;
  if (i < n) p[i] = 0.f;
}

// Fused conv1(2x2,s4)+BN+ReLU -> conv2(1x1,s2)+BN+ReLU -> conv3(1x1,s2)+BN+ReLU -> spatial mean.
// Only the 14x14 positions that survive the stride chain are computed.
// One 128-thread block per (n, p, q); stages ping through LDS; result atomically
// accumulated (pre-scaled by 1/196) into z2f[n, 0..127].
__global__ __launch_bounds__(128) void conv_stack_kern(
    const float* __restrict__ z2,
    const float* __restrict__ w1, const float* __restrict__ w2, const float* __restrict__ w3,
    const float* __restrict__ g1, const float* __restrict__ b1,
    const float* __restrict__ m1, const float* __restrict__ var1,
    const float* __restrict__ g2, const float* __restrict__ b2,
    const float* __restrict__ m2, const float* __restrict__ var2,
    const float* __restrict__ g3, const float* __restrict__ b3,
    const float* __restrict__ m3, const float* __restrict__ var3,
    float* __restrict__ z2f)
{
  __shared__ float zin[12];   // z2 patch: [ic][dy][dx]
  __shared__ float a1[32];    // conv1+bn+relu outputs
  __shared__ float a2[64];    // conv2+bn+relu outputs

  int n  = blockIdx.x / 196;
  int pq = blockIdx.x % 196;
  int p = pq / 14, q = pq % 14;
  int t = threadIdx.x;

  if (t < 12) {
    int ic = t >> 2, dy = (t >> 1) & 1, dx = t & 1;
    zin[t] = z2[(((size_t)n * 3 + ic) * 224 + (16 * p + dy)) * 224 + (16 * q + dx)];
  }
  __syncthreads();

  if (t < 32) {
    float acc = 0.f;
    #pragma unroll
    for (int k = 0; k < 12; ++k) acc = fmaf(w1[t * 12 + k], zin[k], acc);
    float s = g1[t] * rsqrtf(var1[t] + BN_EPS);
    a1[t] = fmaxf((acc - m1[t]) * s + b1[t], 0.f);
  }
  __syncthreads();

  if (t < 64) {
    float acc = 0.f;
    #pragma unroll
    for (int k = 0; k < 32; ++k) acc = fmaf(w2[t * 32 + k], a1[k], acc);
    float s = g2[t] * rsqrtf(var2[t] + BN_EPS);
    a2[t] = fmaxf((acc - m2[t]) * s + b2[t], 0.f);
  }
  __syncthreads();

  {
    float acc = 0.f;
    #pragma unroll 8
    for (int k = 0; k < 64; ++k) acc = fmaf(w3[t * 64 + k], a2[k], acc);
    float s = g3[t] * rsqrtf(var3[t] + BN_EPS);
    float y = fmaxf((acc - m3[t]) * s + b3[t], 0.f);
    atomicAdd(&z2f[n * 128 + t], y * (1.f / 196.f));
  }
}

// C[M,N] = A[M,K] * B[N,K]^T, full f32 via V_WMMA_F32_16X16X4_F32.
// One wave (wave32) computes a 32x32 tile as a 2x2 grid of 16x16 WMMA tiles:
// per k-step, 2 A-loads + 2 B-loads feed 4 WMMAs (halves vmem per matrix op
// vs a 16x16 tile).
// A-operand layout: lane L holds row (row0 + L%16), k = 2*(L/16)+{0,1}  -> contiguous float2
// B-operand layout: lane L holds col (col0 + L%16), k = 2*(L/16)+{0,1}  -> contiguous float2
// D layout: vgpr v: M = row0 + v + 8*(L/16), N = col0 + L%16
__global__ __launch_bounds__(256) void wmma_gemm_nt_kern(
    const float* __restrict__ A, int lda,
    const float* __restrict__ B, int ldb,
    float* __restrict__ C, int ldc,
    int K, int tiles_n2)
{
  int wave = blockIdx.x * (blockDim.x >> 5) + (threadIdx.x >> 5);
  int lane = threadIdx.x & 31;
  int ti = wave / tiles_n2;   // 32-row tile index
  int tn = wave % tiles_n2;   // 32-col tile index
  int hi = lane >> 4;         // 0: k pair {0,1}, 1: k pair {2,3}
  int r  = lane & 15;

  const float* a0 = A + (size_t)(ti * 32 + r)      * lda + 2 * hi;
  const float* a1 = A + (size_t)(ti * 32 + 16 + r) * lda + 2 * hi;
  const float* b0 = B + (size_t)(tn * 32 + r)      * ldb + 2 * hi;
  const float* b1 = B + (size_t)(tn * 32 + 16 + r) * ldb + 2 * hi;

  v8f c00 = {}, c01 = {}, c10 = {}, c11 = {};
  #pragma unroll 2
  for (int kk = 0; kk < K; kk += 4) {
    v2f av0 = *(const v2f*)(a0 + kk);
    v2f av1 = *(const v2f*)(a1 + kk);
    v2f bv0 = *(const v2f*)(b0 + kk);
    v2f bv1 = *(const v2f*)(b1 + kk);
    // 8 args: (neg_a, A, neg_b, B, c_mod, C, reuse_a, reuse_b)
    c00 = __builtin_amdgcn_wmma_f32_16x16x4_f32(false, av0, false, bv0, (short)0, c00, false, false);
    c01 = __builtin_amdgcn_wmma_f32_16x16x4_f32(false, av0, false, bv1, (short)0, c01, false, false);
    c10 = __builtin_amdgcn_wmma_f32_16x16x4_f32(false, av1, false, bv0, (short)0, c10, false, false);
    c11 = __builtin_amdgcn_wmma_f32_16x16x4_f32(false, av1, false, bv1, (short)0, c11, false, false);
  }

  int row0 = ti * 32 + 8 * hi;
  int col0 = tn * 32 + r;
  float* cp00 = C + (size_t)row0 * ldc + col0;
  #pragma unroll
  for (int v = 0; v < 8; ++v) {
    cp00[(size_t)v * ldc]                        = c00[v];
    cp00[(size_t)v * ldc + 16]                   = c01[v];
    cp00[(size_t)(v + 16) * ldc]                 = c10[v];
    cp00[(size_t)(v + 16) * ldc + 16]            = c11[v];
  }
}

// S[i,m] = sum_j relu(h1[i,m] + h2[j,m] + fc1_b[m]).
// Block = (i, 256-wide m chunk); h2 (1MB) stays hot in L2; all reads coalesced.
__global__ __launch_bounds__(256) void sumrelu_kern(
    const float* __restrict__ h1, const float* __restrict__ h2,
    const float* __restrict__ fc1_b, float* __restrict__ S)
{
  int i = blockIdx.x >> 2;
  int m = ((blockIdx.x & 3) << 8) + threadIdx.x;
  float base = h1[i * 1024 + m] + fc1_b[m];
  float acc = 0.f;
  #pragma unroll 8
  for (int j = 0; j < 256; ++j)
    acc += fmaxf(base + h2[j * 1024 + m], 0.f);
  S[i * 1024 + m] = acc;
}

// out[i,c] = sum_m S[i,m] * fc2_w[c,m] + 256 * fc2_b[c]   (256 x 65, K=1024)
__global__ __launch_bounds__(256) void final_gemm_kern(
    const float* __restrict__ S, const float* __restrict__ fc2_w,
    const float* __restrict__ fc2_b, float* __restrict__ out)
{
  int idx = blockIdx.x * blockDim.x + threadIdx.x;
  if (idx >= 256 * 65) return;
  int i = idx / 65, c = idx % 65;
  const float* sp = S + (size_t)i * 1024;
  const float* wp = fc2_w + (size_t)c * 1024;
  float acc = 0.f;
  #pragma unroll 4
  for (int m = 0; m < 1024; m += 4) {
    acc = fmaf(sp[m + 0], wp[m + 0], acc);
    acc = fmaf(sp[m + 1], wp[m + 1], acc);
    acc = fmaf(sp[m + 2], wp[m + 2], acc);
    acc = fmaf(sp[m + 3], wp[m + 3], acc);
  }
  out[idx] = acc + 256.f * fc2_b[c];
}

extern "C" void kernel_launch(void* const* d_in, const int* in_sizes, int n_in,
                              void* d_out, int out_size, void* d_ws, size_t ws_size,
                              hipStream_t stream) {
  const float* z1      = (const float*)d_in[0];   // (256, 2048)
  const float* z2      = (const float*)d_in[1];   // (256, 3, 224, 224)
  const float* conv1_w = (const float*)d_in[2];   // (32, 3, 2, 2)
  const float* conv2_w = (const float*)d_in[3];   // (64, 32, 1, 1)
  const float* conv3_w = (const float*)d_in[4];   // (128, 64, 1, 1)
  const float* bn1_g   = (const float*)d_in[5];
  const float* bn1_b   = (const float*)d_in[6];
  const float* bn1_m   = (const float*)d_in[7];
  const float* bn1_v   = (const float*)d_in[8];
  const float* bn2_g   = (const float*)d_in[9];
  const float* bn2_b   = (const float*)d_in[10];
  const float* bn2_m   = (const float*)d_in[11];
  const float* bn2_v   = (const float*)d_in[12];
  const float* bn3_g   = (const float*)d_in[13];
  const float* bn3_b   = (const float*)d_in[14];
  const float* bn3_m   = (const float*)d_in[15];
  const float* bn3_v   = (const float*)d_in[16];
  const float* fc1_w   = (const float*)d_in[17];  // (1024, 2176)
  const float* fc1_b   = (const float*)d_in[18];  // (1024,)
  const float* fc2_w   = (const float*)d_in[19];  // (65, 1024)
  const float* fc2_b   = (const float*)d_in[20];  // (65,)
  float* out = (float*)d_out;                     // (256, 65)

  float* ws  = (float*)d_ws;
  float* z2f = ws;             // 256*128   = 32768 floats
  float* h1  = z2f + 32768;    // 256*1024  = 262144 floats
  float* h2  = h1 + 262144;    // 256*1024
  float* S   = h2 + 262144;    // 256*1024

  // 1) z2f must be zero for the atomic mean accumulation
  zero_f32_kern<<<128, 256, 0, stream>>>(z2f, 32768);

  // 2) fused conv stack -> z2f (one block per surviving spatial position)
  conv_stack_kern<<<256 * 196, 128, 0, stream>>>(
      z2, conv1_w, conv2_w, conv3_w,
      bn1_g, bn1_b, bn1_m, bn1_v,
      bn2_g, bn2_b, bn2_m, bn2_v,
      bn3_g, bn3_b, bn3_m, bn3_v,
      z2f);

  // 3) h1 = z1 @ w1a^T : M=256, N=1024, K=2048 ; w1a = fc1_w[:, :2048], ldb=2176
  //    (256/32)*(1024/32) = 8*32 = 256 waves -> 32 blocks of 8 waves
  wmma_gemm_nt_kern<<<32, 256, 0, stream>>>(z1, 2048, fc1_w, 2176, h1, 1024, 2048, 32);

  // 4) h2 = z2f @ w1b^T : M=256, N=1024, K=128 ; w1b = fc1_w[:, 2048:], ldb=2176
  wmma_gemm_nt_kern<<<32, 256, 0, stream>>>(z2f, 128, fc1_w + 2048, 2176, h2, 1024, 128, 32);

  // 5) S[i,m] = sum_j relu(h1[i,m] + h2[j,m] + fc1_b[m])
  sumrelu_kern<<<1024, 256, 0, stream>>>(h1, h2, fc1_b, S);

  // 6) out = S @ fc2_w^T + 256*fc2_b
  final_gemm_kern<<<(256 * 65 + 255) / 256, 256, 0, stream>>>(S, fc2_w, fc2_b, out);
}